// GCL_35493609734858
// MI455X (gfx1250) — compile-verified
//
#include <hip/hip_runtime.h>

#define N_TOT   12288
#define D       256
#define NUM     4096
#define ID_BLK  16                 // G rows per workgroup
#define NWAVE   2
#define BLOCK   (NWAVE * 32)
#define GSTR    260                // padded LDS stride for G rows (floats)
#define ASTR    132                // padded LDS stride for A K-chunks (floats)
#define NTILE_IT (N_TOT / 16 / NWAVE)   // 384 column tiles per wave
#define NSTAGES  (NTILE_IT * 2)         // 2 K-chunks (128 each) per tile
#define NEG_CNT 12276.0f           // N - 3*NUM_POS
#define THRESH  1e-6f

typedef __attribute__((ext_vector_type(2))) float v2f;
typedef __attribute__((ext_vector_type(8))) float v8f;

// ---------------------------------------------------------------- row norms
__global__ __launch_bounds__(256) void tl_norms(const float* __restrict__ inp,
                                                float* __restrict__ norms) {
    const int wid  = threadIdx.x >> 5;
    const int lane = threadIdx.x & 31;
    const int row  = blockIdx.x * 8 + wid;          // 12288 / 8 = 1536 blocks
    const float4* p = (const float4*)(inp + (size_t)row * D + lane * 8);
    float4 a = p[0], b = p[1];
    float s = a.x*a.x + a.y*a.y + a.z*a.z + a.w*a.w
            + b.x*b.x + b.y*b.y + b.z*b.z + b.w*b.w;
    for (int off = 16; off > 0; off >>= 1) s += __shfl_xor(s, off, 32);
    if (lane == 0) norms[row] = s;
}

// ---------------------------------------------------------------- fused GEMM + masked reductions
// phase 1: an_rowsum[g] = sum over negative cols of dist; per-WG positive sums/counts
// phase 2: rowmean[g]   = mean of an where (THRESH < an < d_neg[g])
__global__ __launch_bounds__(BLOCK) void tl_pass(
    const float* __restrict__ inp, const int* __restrict__ targets,
    const float* __restrict__ norms,
    float* __restrict__ an_rowsum,
    float* __restrict__ apS, float* __restrict__ apC,
    float* __restrict__ rowmean, int phase) {

    __shared__ __attribute__((aligned(16))) float sg[ID_BLK * GSTR];
    __shared__ __attribute__((aligned(16))) float sa[NWAVE][2][16 * ASTR];
    __shared__ float red_s[NWAVE][32][8];
    __shared__ float red_c[NWAVE][32][8];
    __shared__ float apred[NWAVE][32][2];

    const int tid   = threadIdx.x;
    const int wid   = tid >> 5;
    const int lane  = tid & 31;
    const int lo    = lane & 15;
    const int half  = lane >> 4;
    const int koff2 = half * 2;
    const int gblk  = blockIdx.x;
    const int g0    = gblk * ID_BLK;                // first G row index (0..4095)

    // Stage the 16 G rows (global rows NUM+g0 .. NUM+g0+15) into LDS (once)
    for (int i = tid; i < ID_BLK * (D / 4); i += BLOCK) {
        int r = i >> 6, c4 = (i & 63) << 2;
        float4 v = *(const float4*)(inp + (size_t)(NUM + g0 + r) * D + c4);
        *(float4*)(&sg[r * GSTR + c4]) = v;
    }
    __syncthreads();

    // Per-lane constants for the 8 output rows this lane holds (m = i + 8*half)
    float ng[8]; int tg[8]; float dneg[8];
    #pragma unroll
    for (int i = 0; i < 8; ++i) {
        int g = g0 + i + 8 * half;
        ng[i]   = norms[NUM + g];
        tg[i]   = targets[NUM + g];
        dneg[i] = (phase == 2) ? (an_rowsum[g] * (1.0f / NEG_CNT)) : 0.0f;
    }

    float acc_s[8], acc_c[8];
    #pragma unroll
    for (int i = 0; i < 8; ++i) { acc_s[i] = 0.0f; acc_c[i] = 0.0f; }
    float apsum = 0.0f, apcnt = 0.0f;

    // LDS byte offsets of the two per-wave async staging buffers
    const unsigned abase[2] = { (unsigned)(uintptr_t)&sa[wid][0][0],
                                (unsigned)(uintptr_t)&sa[wid][1][0] };

    // Issue one stage (16 rows x 128 K-values) of async global->LDS b128 loads.
    // Stage s: tile it = s>>1 (columns nb..nb+15), K-chunk kh = s&1 ([kh*128, kh*128+128)).
    auto issue_stage = [&](int s, int b) {
        const int it = s >> 1, kh = s & 1;
        const int nb = (it * NWAVE + wid) * 16;
        #pragma unroll
        for (int i = 0; i < 16; ++i) {
            int idx = i * 32 + lane;                 // 512 float4 per stage
            int r = idx >> 5, c4 = (idx & 31) << 2;  // c4: float index within chunk
            unsigned ldsoff = abase[b] + (unsigned)((r * ASTR + c4) * 4);
            unsigned goff   = (unsigned)(((nb + r) * D + kh * 128 + c4) * 4);
            asm volatile("global_load_async_to_lds_b128 %0, %1, %2 offset:0"
                         :: "v"(ldsoff), "v"(goff), "s"(inp) : "memory");
        }
    };

    issue_stage(0, 0);
    int cur = 0;
    v8f c = {};

    for (int s = 0; s < NSTAGES; ++s) {
        if (s + 1 < NSTAGES) {
            issue_stage(s + 1, cur ^ 1);
            asm volatile("s_wait_asynccnt 0x10" ::: "memory");  // stage s landed
        } else {
            asm volatile("s_wait_asynccnt 0x0" ::: "memory");
        }

        const float* saw = &sa[wid][cur][0];
        const int kg0 = (s & 1) * 32;               // global K-step base (steps of 4)
        #pragma unroll 8
        for (int k = 0; k < 32; ++k) {
            v2f a = *(const v2f*)(&sg [lo * GSTR + (kg0 + k) * 4 + koff2]);
            v2f b = *(const v2f*)(&saw[lo * ASTR +  k        * 4 + koff2]);
            c = __builtin_amdgcn_wmma_f32_16x16x4_f32(false, a, false, b,
                                                      (short)0, c, false, false);
        }

        if (s & 1) {
            // Tile complete: c[i] = dot(G[row], A[col]), row = i + 8*half, col = nb+lo
            const int   it = s >> 1;
            const int   n  = (it * NWAVE + wid) * 16 + lo;
            const float na = norms[n];
            const int   tc = targets[n];
            #pragma unroll
            for (int i = 0; i < 8; ++i) {
                float d2   = ng[i] + na - 2.0f * c[i];
                float dist = __builtin_amdgcn_sqrtf(fmaxf(d2, 1e-12f)); // raw v_sqrt_f32
                bool  neg  = (tg[i] != tc);
                if (phase == 1) {
                    if (neg)                  acc_s[i] += dist;
                    else if (dist > THRESH) { apsum += dist; apcnt += 1.0f; }
                } else {
                    if (neg && dist > THRESH && dist < dneg[i]) {
                        acc_s[i] += dist; acc_c[i] += 1.0f;
                    }
                }
            }
            c = (v8f){};
        }
        cur ^= 1;
    }

    // Deterministic block-level reduction through LDS (no float atomics)
    #pragma unroll
    for (int i = 0; i < 8; ++i) { red_s[wid][lane][i] = acc_s[i]; red_c[wid][lane][i] = acc_c[i]; }
    apred[wid][lane][0] = apsum; apred[wid][lane][1] = apcnt;
    __syncthreads();

    if (tid < 16) {
        int r = tid, h = r >> 3, i = r & 7;
        float s = 0.0f, cnt = 0.0f;
        for (int w = 0; w < NWAVE; ++w)
            for (int l = 0; l < 16; ++l) { s += red_s[w][h * 16 + l][i]; cnt += red_c[w][h * 16 + l][i]; }
        if (phase == 1) an_rowsum[g0 + r] = s;
        else            rowmean [g0 + r] = s / cnt;
    }
    if (phase == 1 && tid == 0) {
        float s = 0.0f, c2 = 0.0f;
        for (int w = 0; w < NWAVE; ++w)
            for (int l = 0; l < 32; ++l) { s += apred[w][l][0]; c2 += apred[w][l][1]; }
        apS[gblk] = s; apC[gblk] = c2;
    }
}

// ---------------------------------------------------------------- final scalar
__global__ __launch_bounds__(256) void tl_finalize(const float* __restrict__ rowmean,
                                                   const float* __restrict__ apS,
                                                   const float* __restrict__ apC,
                                                   float* __restrict__ out) {
    __shared__ float s1[256], s2[256], s3[256];
    const int t = threadIdx.x;
    float a = 0.0f;
    for (int i = t; i < NUM; i += 256) a += rowmean[i];
    s1[t] = a; s2[t] = apS[t]; s3[t] = apC[t];
    __syncthreads();
    for (int off = 128; off > 0; off >>= 1) {
        if (t < off) { s1[t] += s1[t + off]; s2[t] += s2[t + off]; s3[t] += s3[t + off]; }
        __syncthreads();
    }
    if (t == 0) {
        float an_mean = s1[0] * (1.0f / (float)NUM);
        float ap_mean = s2[0] / s3[0];
        out[0] = ap_mean / an_mean;
    }
}

// ---------------------------------------------------------------- launcher
extern "C" void kernel_launch(void* const* d_in, const int* in_sizes, int n_in,
                              void* d_out, int out_size, void* d_ws, size_t ws_size,
                              hipStream_t stream) {
    const float* inp     = (const float*)d_in[0];   // (12288, 256) fp32
    const int*   targets = (const int*)d_in[1];     // (12288,) int

    float* ws        = (float*)d_ws;
    float* norms     = ws;                          // 12288
    float* an_rowsum = ws + 12288;                  // 4096
    float* apS       = an_rowsum + 4096;            // 256 (per-WG positive sums)
    float* apC       = apS + 256;                   // 256 (per-WG positive counts)
    float* rowmean   = apC + 256;                   // 4096

    tl_norms   <<<N_TOT / 8, 256, 0, stream>>>(inp, norms);
    tl_pass    <<<NUM / ID_BLK, BLOCK, 0, stream>>>(inp, targets, norms,
                                                    an_rowsum, apS, apC, rowmean, 1);
    tl_pass    <<<NUM / ID_BLK, BLOCK, 0, stream>>>(inp, targets, norms,
                                                    an_rowsum, apS, apC, rowmean, 2);
    tl_finalize<<<1, 256, 0, stream>>>(rowmean, apS, apC, (float*)d_out);
}